// WassersteinGaussian_80771154969000
// MI455X (gfx1250) — compile-verified
//
#include <hip/hip_runtime.h>
#include <stdint.h>
#include <stddef.h>

#define TPB 256

// CDNA5 async DMA: global -> LDS, 16 bytes per lane, tracked by ASYNCcnt.
// GVS addressing mode: addr = SGPR64 base + VGPR32 offset (+ inst offset).
// VDST VGPR holds the workgroup-relative LDS byte offset.
// th:TH_LOAD_NT — streaming data (260 MB, read once) must not thrash L2 (192 MB).
__device__ __forceinline__ void async_copy16(uint32_t lds_off, uint32_t voff, uint64_t sbase) {
    asm volatile("global_load_async_to_lds_b128 %0, %1, %2 th:TH_LOAD_NT"
                 :: "v"(lds_off), "v"(voff), "s"(sbase)
                 : "memory");
}

__device__ __forceinline__ void wait_async_all() {
    asm volatile("s_wait_asynccnt 0" ::: "memory");
}

// LDS word-offsets for one block's 256-element AoS chunk (30720 bytes total)
//  loc1   @ 0     (768 floats)
//  scale1 @ 768   (768 floats)
//  rot1   @ 1536  (2304 floats)
//  loc2   @ 3840  (768 floats)
//  scale2 @ 4608  (768 floats)
//  rot2   @ 5376  (2304 floats)
__global__ __launch_bounds__(TPB) void wasserstein_gauss_kernel(
    const float* __restrict__ loc1, const float* __restrict__ scale1,
    const float* __restrict__ rot1, const float* __restrict__ loc2,
    const float* __restrict__ scale2, const float* __restrict__ rot2,
    float* __restrict__ out, int nB)
{
    __shared__ float smem[7680];

    const int tid  = threadIdx.x;
    const int base = blockIdx.x * TPB;
    const int i    = base + tid;

    float l1[3], s1[3], r1[9], l2[3], s2[3], r2[9];

    if (base + TPB <= nB) {
        // ---- stage this block's chunk into LDS via async b128 DMA (GVS mode) ----
        const uint32_t sb   = (uint32_t)(uintptr_t)(void*)smem;
        const uint64_t g_l1 = (uint64_t)(uintptr_t)(loc1   + (size_t)base * 3);
        const uint64_t g_s1 = (uint64_t)(uintptr_t)(scale1 + (size_t)base * 3);
        const uint64_t g_r1 = (uint64_t)(uintptr_t)(rot1   + (size_t)base * 9);
        const uint64_t g_l2 = (uint64_t)(uintptr_t)(loc2   + (size_t)base * 3);
        const uint64_t g_s2 = (uint64_t)(uintptr_t)(scale2 + (size_t)base * 3);
        const uint64_t g_r2 = (uint64_t)(uintptr_t)(rot2   + (size_t)base * 9);

        // loc/scale chunks: 3072 B each = 192 x b128 (waves 0..5 active, 1 iter)
        for (int j = tid; j < 192; j += TPB) {
            const uint32_t o = (uint32_t)j * 16u;
            async_copy16(sb +        0u + o, o, g_l1);
            async_copy16(sb +  3072u + o, o, g_s1);
            async_copy16(sb + 15360u + o, o, g_l2);
            async_copy16(sb + 18432u + o, o, g_s2);
        }
        // rot chunks: 9216 B each = 576 x b128
        for (int j = tid; j < 576; j += TPB) {
            const uint32_t o = (uint32_t)j * 16u;
            async_copy16(sb +  6144u + o, o, g_r1);
            async_copy16(sb + 21504u + o, o, g_r2);
        }
        wait_async_all();
        __syncthreads();

        // conflict-free LDS reads (strides 3 and 9 words are odd)
        #pragma unroll
        for (int k = 0; k < 3; ++k) {
            l1[k] = smem[        3 * tid + k];
            s1[k] = smem[ 768 +  3 * tid + k];
            l2[k] = smem[3840 +  3 * tid + k];
            s2[k] = smem[4608 +  3 * tid + k];
        }
        #pragma unroll
        for (int k = 0; k < 9; ++k) {
            r1[k] = smem[1536 + 9 * tid + k];
            r2[k] = smem[5376 + 9 * tid + k];
        }
    } else {
        if (i >= nB) return;
        #pragma unroll
        for (int k = 0; k < 3; ++k) {
            l1[k] = loc1[3 * i + k];
            s1[k] = scale1[3 * i + k];
            l2[k] = loc2[3 * i + k];
            s2[k] = scale2[3 * i + k];
        }
        #pragma unroll
        for (int k = 0; k < 9; ++k) {
            r1[k] = rot1[9 * i + k];
            r2[k] = rot2[9 * i + k];
        }
    }

    // ---- ||mu1 - mu2||^2 ----
    const float dx = l1[0] - l2[0], dy = l1[1] - l2[1], dz = l1[2] - l2[2];
    const float loc_diff2 = dx * dx + dy * dy + dz * dz;

    // ---- cov2 = R2 diag(s2) R2^T  (symmetric, 6 unique entries) ----
    const float s2a = s2[0], s2b = s2[1], s2c = s2[2];
    const float c00 = s2a*r2[0]*r2[0] + s2b*r2[1]*r2[1] + s2c*r2[2]*r2[2];
    const float c01 = s2a*r2[0]*r2[3] + s2b*r2[1]*r2[4] + s2c*r2[2]*r2[5];
    const float c02 = s2a*r2[0]*r2[6] + s2b*r2[1]*r2[7] + s2c*r2[2]*r2[8];
    const float c11 = s2a*r2[3]*r2[3] + s2b*r2[4]*r2[4] + s2c*r2[5]*r2[5];
    const float c12 = s2a*r2[3]*r2[6] + s2b*r2[4]*r2[7] + s2c*r2[5]*r2[8];
    const float c22 = s2a*r2[6]*r2[6] + s2b*r2[7]*r2[7] + s2c*r2[8]*r2[8];

    // ---- T = cov2 * R1 ----
    const float t00 = c00*r1[0] + c01*r1[3] + c02*r1[6];
    const float t01 = c00*r1[1] + c01*r1[4] + c02*r1[7];
    const float t02 = c00*r1[2] + c01*r1[5] + c02*r1[8];
    const float t10 = c01*r1[0] + c11*r1[3] + c12*r1[6];
    const float t11 = c01*r1[1] + c11*r1[4] + c12*r1[7];
    const float t12 = c01*r1[2] + c11*r1[5] + c12*r1[8];
    const float t20 = c02*r1[0] + c12*r1[3] + c22*r1[6];
    const float t21 = c02*r1[1] + c12*r1[4] + c22*r1[7];
    const float t22 = c02*r1[2] + c12*r1[5] + c22*r1[8];

    // ---- M = R1^T * T ----
    const float m00 = r1[0]*t00 + r1[3]*t10 + r1[6]*t20;
    const float m01 = r1[0]*t01 + r1[3]*t11 + r1[6]*t21;
    const float m02 = r1[0]*t02 + r1[3]*t12 + r1[6]*t22;
    const float m10 = r1[1]*t00 + r1[4]*t10 + r1[7]*t20;
    const float m11 = r1[1]*t01 + r1[4]*t11 + r1[7]*t21;
    const float m12 = r1[1]*t02 + r1[4]*t12 + r1[7]*t22;
    const float m20 = r1[2]*t00 + r1[5]*t10 + r1[8]*t20;
    const float m21 = r1[2]*t01 + r1[5]*t11 + r1[8]*t21;
    const float m22 = r1[2]*t02 + r1[5]*t12 + r1[8]*t22;
    const float sm01 = 0.5f * (m01 + m10);
    const float sm02 = 0.5f * (m02 + m20);
    const float sm12 = 0.5f * (m12 + m21);

    // ---- E = diag(sqrt(s1)) M diag(sqrt(s1)) ----
    const float s1a = s1[0], s1b = s1[1], s1c = s1[2];
    const float q0 = sqrtf(s1a), q1 = sqrtf(s1b), q2 = sqrtf(s1c);
    const float e00 = s1a * m00;
    const float e11 = s1b * m11;
    const float e22 = s1c * m22;
    const float e01 = q0 * q1 * sm01;
    const float e02 = q0 * q2 * sm02;
    const float e12 = q1 * q2 * sm12;

    // ---- eigenvalues of symmetric 3x3 (trigonometric closed form) ----
    const float p1 = e01*e01 + e02*e02 + e12*e12;
    const float q  = (e00 + e11 + e22) * (1.0f / 3.0f);
    const float b00 = e00 - q, b11 = e11 - q, b22 = e22 - q;
    const float p2 = b00*b00 + b11*b11 + b22*b22 + 2.0f * p1;
    const float p  = sqrtf(p2 * (1.0f / 6.0f));
    const float inv_p = (p > 1e-20f) ? (1.0f / p) : 0.0f;

    const float detB = b00 * (b11 * b22 - e12 * e12)
                     - e01 * (e01 * b22 - e12 * e02)
                     + e02 * (e01 * e12 - b11 * e02);
    float r = 0.5f * detB * inv_p * inv_p * inv_p;
    r = fminf(1.0f, fmaxf(-1.0f, r));

    const float phi = acosf(r) * (1.0f / 3.0f);
    const float eig1 = q + 2.0f * p * cosf(phi);
    const float eig3 = q + 2.0f * p * cosf(phi + 2.09439510239319549f); // + 2*pi/3
    const float eig2 = 3.0f * q - eig1 - eig3;

    // sum sqrt(|lambda|)  ( == sum (lambda^2)^(1/4) )
    const float tr_sqrt = sqrtf(fabsf(eig1)) + sqrtf(fabsf(eig2)) + sqrtf(fabsf(eig3));

    float cov_w = (s1a + s1b + s1c) + (s2a + s2b + s2c) - 2.0f * tr_sqrt;
    cov_w = fmaxf(cov_w, 0.0f);

    // one-pass output: non-temporal store
    __builtin_nontemporal_store(sqrtf(loc_diff2 + cov_w), &out[i]);
}

extern "C" void kernel_launch(void* const* d_in, const int* in_sizes, int n_in,
                              void* d_out, int out_size, void* d_ws, size_t ws_size,
                              hipStream_t stream) {
    (void)n_in; (void)out_size; (void)d_ws; (void)ws_size;
    const float* loc1   = (const float*)d_in[0];
    const float* scale1 = (const float*)d_in[1];
    const float* rot1   = (const float*)d_in[2];
    const float* loc2   = (const float*)d_in[3];
    const float* scale2 = (const float*)d_in[4];
    const float* rot2   = (const float*)d_in[5];
    float* out = (float*)d_out;

    const int nB = in_sizes[0] / 3;
    const int blocks = (nB + TPB - 1) / TPB;
    hipLaunchKernelGGL(wasserstein_gauss_kernel, dim3(blocks), dim3(TPB), 0, stream,
                       loc1, scale1, rot1, loc2, scale2, rot2, out, nB);
}